// DMM_70927089926358
// MI455X (gfx1250) — compile-verified
//
#include <hip/hip_runtime.h>
#include <hip/hip_bf16.h>

// ---- problem dims (match reference) ----
static constexpr int NB  = 512;   // batch
static constexpr int NT  = 256;   // time
static constexpr int DIN = 88;
static constexpr int ZD  = 64;
static constexpr int ED  = 100;
static constexpr int TRD = 200;
static constexpr int HD  = 600;
static constexpr float LOG2PI_F = 1.8378770664093453f;

typedef float v2f __attribute__((ext_vector_type(2)));
typedef float v8f __attribute__((ext_vector_type(8)));

// ---------------- WMMA fp32 16x16x4 helpers ----------------
// A (16x4): lane<16 -> m=lane, {K=k0,k0+1}; lane>=16 -> m=lane-16, {K=k0+2,k0+3}
__device__ __forceinline__ v2f ldA(const float* __restrict__ base, int row0,
                                   long stride, int k0, int lane) {
  int m = lane & 15;
  int kk = k0 + ((lane >> 4) << 1);
  const float* p = base + (long)(row0 + m) * stride + kk;
  v2f a; a.x = p[0]; a.y = p[1];
  return a;
}
// B (4x16) where B[k][n] = W[n][k]; W row-major [nfeat x kfeat].
// Out-of-range n lanes CLAMP to the last valid row (branch-free); the resulting
// duplicate C columns are never stored (epilogue guards n < nout).
__device__ __forceinline__ v2f ldBT(const float* __restrict__ W, int n0, int nlim,
                                    int kfeat, int k0, int lane) {
  int n = n0 + (lane & 15);
  n = (n < nlim) ? n : (nlim - 1);
  int kk = k0 + ((lane >> 4) << 1);
  const float* p = W + (long)n * kfeat + kk;
  v2f b; b.x = p[0]; b.y = p[1];
  return b;
}
__device__ __forceinline__ v2f ldA_lds(const float* s, int stride, int k0, int lane) {
  int m = lane & 15;
  int kk = k0 + ((lane >> 4) << 1);
  const float* p = s + m * stride + kk;
  v2f a; a.x = p[0]; a.y = p[1];
  return a;
}
__device__ __forceinline__ v2f ldA_lds_relu(const float* s, int stride, int k0, int lane) {
  int m = lane & 15;
  int kk = k0 + ((lane >> 4) << 1);
  const float* p = s + m * stride + kk;
  v2f a; a.x = fmaxf(p[0], 0.f); a.y = fmaxf(p[1], 0.f);
  return a;
}
__device__ __forceinline__ v8f wmma4(v2f a, v2f b, v8f c) {
  return __builtin_amdgcn_wmma_f32_16x16x4_f32(false, a, false, b, (short)0, c, false, false);
}

__device__ __forceinline__ float softplusf(float x) {
  return (x > 20.f) ? x : log1pf(expf(x));
}
__device__ __forceinline__ float logsigmoidf(float x) {
  return (x >= 0.f) ? -log1pf(expf(-x)) : (x - log1pf(expf(x)));
}

// C[16 x nout] = act(A_lds[16 x kdim] @ W^T + bias) -> out_lds (stride ostride)
// act: 0 none, 1 relu, 2 sigmoid, 3 softplus
__device__ __forceinline__ void gemm_lds(const float* a_lds, int astride, int kdim,
                                         const float* __restrict__ W,
                                         const float* __restrict__ bias, int nout,
                                         float* out_lds, int ostride, int act,
                                         int lane, bool relu_a) {
  int half = lane >> 4, ml = lane & 15;
  for (int n0 = 0; n0 < nout; n0 += 16) {
    v8f c = {};
    for (int k = 0; k < kdim; k += 4) {
      v2f a = relu_a ? ldA_lds_relu(a_lds, astride, k, lane)
                     : ldA_lds(a_lds, astride, k, lane);
      v2f b = ldBT(W, n0, nout, kdim, k, lane);
      c = wmma4(a, b, c);
    }
    int n = n0 + ml;
    if (n < nout) {
      float bn = bias[n];
#pragma unroll
      for (int i = 0; i < 8; ++i) {
        float v = c[i] + bn;
        if (act == 1) v = fmaxf(v, 0.f);
        else if (act == 2) v = 1.f / (1.f + expf(-v));
        else if (act == 3) v = softplusf(v);
        out_lds[(i + 8 * half) * ostride + n] = v;
      }
    }
  }
}

// ---------------- init ----------------
__global__ void init_h_kernel(const float* __restrict__ h0, float* __restrict__ H) {
  int i = blockIdx.x * 256 + threadIdx.x;
  if (i < NB * HD) H[i] = h0[i % HD];
}
__global__ void zero_kernel(float* __restrict__ p, int n) {
  int i = blockIdx.x * 256 + threadIdx.x;
  if (i < n) p[i] = 0.f;
}

// ---------------- RNN step: Hnew = relu(x_t@W_ih^T + b_ih + Hold@W_hh^T + b_hh) ----------------
__global__ void __launch_bounds__(32) rnn_step_kernel(
    const float* __restrict__ x,        // [NB,NT,DIN]
    const float* __restrict__ W_ih, const float* __restrict__ b_ih,
    const float* __restrict__ W_hh, const float* __restrict__ b_hh,
    const float* __restrict__ Hold, float* __restrict__ Hnew, int t) {
  int lane = threadIdx.x & 31;
  int m0 = blockIdx.x * 16;   // batch tile
  int n0 = blockIdx.y * 16;   // H tile
  v8f acc = {};
  const float* xbase = x + (long)t * DIN;       // row stride NT*DIN
  for (int k = 0; k < DIN; k += 4) {
    v2f a = ldA(xbase, m0, (long)NT * DIN, k, lane);
    v2f b = ldBT(W_ih, n0, HD, DIN, k, lane);
    acc = wmma4(a, b, acc);
  }
  for (int k = 0; k < HD; k += 4) {
    v2f a = ldA(Hold, m0, HD, k, lane);
    v2f b = ldBT(W_hh, n0, HD, HD, k, lane);
    acc = wmma4(a, b, acc);
  }
  int half = lane >> 4, ml = lane & 15;
  int n = n0 + ml;
  if (n < HD) {
    float bias = b_ih[n] + b_hh[n];
#pragma unroll
    for (int i = 0; i < 8; ++i) {
      int row = m0 + i + 8 * half;
      float v = acc[i] + bias;
      Hnew[(long)row * HD + n] = v > 0.f ? v : 0.f;
    }
  }
}

// ---------------- combiner: loc/scale, sample z, log_q partial ----------------
__global__ void __launch_bounds__(32) combiner_step_kernel(
    const float* __restrict__ Hcur,
    const float* __restrict__ W_loc, const float* __restrict__ b_loc,
    const float* __restrict__ W_scale, const float* __restrict__ b_scale,
    const float* __restrict__ eps,   // [NB,NT,ZD]
    float* __restrict__ zbuf,        // [NB,NT,ZD]
    float* __restrict__ logq,        // [NB*NT]
    int t) {
  __shared__ float red[16][17];
  int lane = threadIdx.x & 31;
  int m0 = blockIdx.x * 16;   // batch tile
  int n0 = blockIdx.y * 16;   // Z tile
  v8f cl = {}, cs = {};
  for (int k = 0; k < HD; k += 4) {
    v2f a  = ldA(Hcur, m0, HD, k, lane);
    v2f bl = ldBT(W_loc,   n0, ZD, HD, k, lane);
    v2f bs = ldBT(W_scale, n0, ZD, HD, k, lane);
    cl = wmma4(a, bl, cl);
    cs = wmma4(a, bs, cs);
  }
  int half = lane >> 4, ml = lane & 15;
  int n = n0 + ml;
  float bln = b_loc[n], bsn = b_scale[n];
#pragma unroll
  for (int i = 0; i < 8; ++i) {
    int brow = m0 + i + 8 * half;
    long rt = (long)brow * NT + t;
    float loc = cl[i] + bln;
    float sc  = softplusf(cs[i] + bsn);
    float e   = eps[rt * ZD + n];
    zbuf[rt * ZD + n] = loc + sc * e;
    red[i + 8 * half][ml] = -0.5f * e * e - logf(sc) - 0.5f * LOG2PI_F;
  }
  __syncthreads();
  if (lane < 16) {
    float s = 0.f;
#pragma unroll
    for (int j = 0; j < 16; ++j) s += red[lane][j];
    atomicAdd(&logq[(long)(m0 + lane) * NT + t], s);
  }
}

// ---------------- model: gated transition + emitter per 16 rows of (b,t) ----------------
__global__ void __launch_bounds__(32) model_kernel(
    const float* __restrict__ zbuf, const float* __restrict__ z0,
    const float* __restrict__ Wg1, const float* __restrict__ bg1,
    const float* __restrict__ Wg2, const float* __restrict__ bg2,
    const float* __restrict__ Wp1, const float* __restrict__ bp1,
    const float* __restrict__ Wp2, const float* __restrict__ bp2,
    const float* __restrict__ Wsig, const float* __restrict__ bsig,
    const float* __restrict__ Wzloc, const float* __restrict__ bzloc,
    const float* __restrict__ We1, const float* __restrict__ be1,
    const float* __restrict__ We2, const float* __restrict__ be2,
    const float* __restrict__ We3, const float* __restrict__ be3,
    const float* __restrict__ x,   // [NB,NT,DIN]
    float* __restrict__ logpz, float* __restrict__ logpx) {
  __shared__ float sBig0[16 * 208];
  __shared__ float sBig1[16 * 208];
  __shared__ float sZS[16 * 64], sZ[16 * 64], sGate[16 * 64];
  __shared__ float sPM[16 * 64], sLoc[16 * 64], sScl[16 * 64];
  int lane = threadIdx.x;
  int row0 = blockIdx.x * 16;

  // stage z and z_shift (rows within a tile share the same b since NT%16==0)
  for (int idx = lane; idx < 16 * 64; idx += 32) {
    int m = idx >> 6, k = idx & 63;
    long row = row0 + m;
    int t = (int)(row & (NT - 1));
    sZ[idx]  = zbuf[row * ZD + k];
    sZS[idx] = (t == 0) ? z0[k] : zbuf[(row - 1) * ZD + k];
  }
  __syncthreads();

  gemm_lds(sZS, 64, ZD, Wg1, bg1, TRD, sBig0, 208, 1, lane, false);   // g1=relu
  __syncthreads();
  gemm_lds(sBig0, 208, TRD, Wg2, bg2, ZD, sGate, 64, 2, lane, false); // gate=sigmoid
  __syncthreads();
  gemm_lds(sZS, 64, ZD, Wp1, bp1, TRD, sBig0, 208, 1, lane, false);   // p1=relu
  __syncthreads();
  gemm_lds(sBig0, 208, TRD, Wp2, bp2, ZD, sPM, 64, 0, lane, false);   // pm
  __syncthreads();
  gemm_lds(sZS, 64, ZD, Wzloc, bzloc, ZD, sLoc, 64, 0, lane, false);  // zl
  __syncthreads();
  for (int idx = lane; idx < 16 * 64; idx += 32) {
    float g = sGate[idx];
    sLoc[idx] = (1.f - g) * sLoc[idx] + g * sPM[idx];                 // z_loc_p
  }
  __syncthreads();
  gemm_lds(sPM, 64, ZD, Wsig, bsig, ZD, sScl, 64, 3, lane, true);     // softplus(relu(pm)@Wsig^T+b)
  __syncthreads();

  if (lane < 16) {
    long row = row0 + lane;
    float s = 0.f;
    for (int k = 0; k < ZD; ++k) {
      float zv = sZ[lane * 64 + k], loc = sLoc[lane * 64 + k], sc = sScl[lane * 64 + k];
      float d = (zv - loc) / sc;
      s += -0.5f * d * d - logf(sc) - 0.5f * LOG2PI_F;
    }
    logpz[row] = s;
  }
  __syncthreads();

  // emitter
  gemm_lds(sZ, 64, ZD, We1, be1, ED, sBig0, 104, 1, lane, false);     // h1
  __syncthreads();
  gemm_lds(sBig0, 104, ED, We2, be2, ED, sBig1, 104, 1, lane, false); // h2
  __syncthreads();

  // logits + per-element Bernoulli log-prob into sBig0 (stride 96)
  {
    int half = lane >> 4, ml = lane & 15;
    for (int n0 = 0; n0 < DIN; n0 += 16) {
      v8f c = {};
      for (int k = 0; k < ED; k += 4) {
        v2f a = ldA_lds(sBig1, 104, k, lane);
        v2f b = ldBT(We3, n0, DIN, ED, k, lane);
        c = wmma4(a, b, c);
      }
      int n = n0 + ml;
#pragma unroll
      for (int i = 0; i < 8; ++i) {
        int m = i + 8 * half;
        float v = 0.f;
        if (n < DIN) {
          float l = c[i] + be3[n];
          long row = row0 + m;
          float xv = x[row * DIN + n];
          v = xv * logsigmoidf(l) + (1.f - xv) * logsigmoidf(-l);
        }
        sBig0[m * 96 + n] = v;
      }
    }
  }
  __syncthreads();
  if (lane < 16) {
    long row = row0 + lane;
    float s = 0.f;
    for (int n = 0; n < DIN; ++n) s += sBig0[lane * 96 + n];
    logpx[row] = s;
  }
}

// ---------------- final reduction over T ----------------
__global__ void __launch_bounds__(256) reduce_kernel(
    const float* __restrict__ mask, const float* __restrict__ logpz,
    const float* __restrict__ logpx, const float* __restrict__ logq,
    float* __restrict__ out) {
  __shared__ float sred[256];
  int b = blockIdx.x, tid = threadIdx.x;
  long base = (long)b * NT;
  sred[tid] = mask[base + tid] *
              (logpz[base + tid] + logpx[base + tid] - logq[base + tid]);
  __syncthreads();
  for (int s = 128; s > 0; s >>= 1) {
    if (tid < s) sred[tid] += sred[tid + s];
    __syncthreads();
  }
  if (tid == 0) out[b] = sred[0];
}

extern "C" void kernel_launch(void* const* d_in, const int* in_sizes, int n_in,
                              void* d_out, int out_size, void* d_ws, size_t ws_size,
                              hipStream_t stream) {
  const float* x       = (const float*)d_in[0];
  const float* mask    = (const float*)d_in[1];
  const float* eps     = (const float*)d_in[2];
  const float* W_ih    = (const float*)d_in[3];
  const float* b_ih    = (const float*)d_in[4];
  const float* W_hh    = (const float*)d_in[5];
  const float* b_hh    = (const float*)d_in[6];
  const float* h_0     = (const float*)d_in[7];
  const float* W_loc   = (const float*)d_in[8];
  const float* b_loc   = (const float*)d_in[9];
  const float* W_scale = (const float*)d_in[10];
  const float* b_scale = (const float*)d_in[11];
  const float* Wg1 = (const float*)d_in[12]; const float* bg1 = (const float*)d_in[13];
  const float* Wg2 = (const float*)d_in[14]; const float* bg2 = (const float*)d_in[15];
  const float* Wp1 = (const float*)d_in[16]; const float* bp1 = (const float*)d_in[17];
  const float* Wp2 = (const float*)d_in[18]; const float* bp2 = (const float*)d_in[19];
  const float* Wsig  = (const float*)d_in[20]; const float* bsig  = (const float*)d_in[21];
  const float* Wzloc = (const float*)d_in[22]; const float* bzloc = (const float*)d_in[23];
  const float* We1 = (const float*)d_in[24]; const float* be1 = (const float*)d_in[25];
  const float* We2 = (const float*)d_in[26]; const float* be2 = (const float*)d_in[27];
  const float* We3 = (const float*)d_in[28]; const float* be3 = (const float*)d_in[29];
  const float* z_0 = (const float*)d_in[30];
  float* out = (float*)d_out;

  float* ws    = (float*)d_ws;
  float* Hbuf0 = ws;                                  // NB*HD
  float* Hbuf1 = Hbuf0 + (size_t)NB * HD;             // NB*HD
  float* zbuf  = Hbuf1 + (size_t)NB * HD;             // NB*NT*ZD
  float* logq  = zbuf + (size_t)NB * NT * ZD;         // NB*NT
  float* logpz = logq + (size_t)NB * NT;
  float* logpx = logpz + (size_t)NB * NT;

  init_h_kernel<<<(NB * HD + 255) / 256, 256, 0, stream>>>(h_0, Hbuf0);
  zero_kernel<<<(NB * NT + 255) / 256, 256, 0, stream>>>(logq, NB * NT);

  float* Hold = Hbuf0;
  float* Hnew = Hbuf1;
  for (int s = 0; s < NT; ++s) {
    int t = NT - 1 - s;  // scan step s produces rnn_out at time t with input x[:, t]
    rnn_step_kernel<<<dim3(NB / 16, (HD + 15) / 16), 32, 0, stream>>>(
        x, W_ih, b_ih, W_hh, b_hh, Hold, Hnew, t);
    combiner_step_kernel<<<dim3(NB / 16, ZD / 16), 32, 0, stream>>>(
        Hnew, W_loc, b_loc, W_scale, b_scale, eps, zbuf, logq, t);
    float* tmp = Hold; Hold = Hnew; Hnew = tmp;
  }

  model_kernel<<<(NB * NT) / 16, 32, 0, stream>>>(
      zbuf, z_0, Wg1, bg1, Wg2, bg2, Wp1, bp1, Wp2, bp2, Wsig, bsig,
      Wzloc, bzloc, We1, be1, We2, be2, We3, be3, x, logpz, logpx);

  reduce_kernel<<<NB, 256, 0, stream>>>(mask, logpz, logpx, logq, out);
}